// ResidualAttentionBlock_54391465836607
// MI455X (gfx1250) — compile-verified
//
#include <hip/hip_runtime.h>
#include <math.h>

typedef __attribute__((ext_vector_type(2))) float v2f;
typedef __attribute__((ext_vector_type(8))) float v8f;

#define D_MODEL 1024
#define D_FF    4096
#define SEQ     1024
#define BATCH   8
#define NHEAD   16
#define HDIM    64
#define MTOT    (BATCH * SEQ)

// ---------------------------------------------------------------------------
// CDNA5 fp32 WMMA: D(16x16 f32) = A(16x4 f32) * B(4x16 f32) + C
// ---------------------------------------------------------------------------
__device__ __forceinline__ v8f wmma_f32(v2f a, v2f b, v8f c) {
  return __builtin_amdgcn_wmma_f32_16x16x4_f32(false, a, false, b, (short)0, c,
                                               false, false);
}

// ---------------------------------------------------------------------------
// gfx1250 async global->LDS copy (ASYNCcnt-tracked, no VGPR round trip).
// LDS dest = low 32 bits of the generic pointer (LDS aperture keeps the
// group-segment offset in addr[31:0]); INST_OFFSET applies to both addresses.
// ---------------------------------------------------------------------------
__device__ __forceinline__ unsigned lds_addr(const void* p) {
  return (unsigned)(uintptr_t)p;
}
__device__ __forceinline__ void async_ld16(unsigned lds, const void* g) {
  asm volatile("global_load_async_to_lds_b128 %0, %1, off" ::"v"(lds), "v"(g)
               : "memory");
}
__device__ __forceinline__ void async_ld16_o16(unsigned lds, const void* g) {
  asm volatile("global_load_async_to_lds_b128 %0, %1, off offset:16" ::"v"(lds),
                   "v"(g)
               : "memory");
}
__device__ __forceinline__ void wait_async0() {
  asm volatile("s_wait_asynccnt 0x0" ::: "memory");
}
__device__ __forceinline__ void wait_async2() {
  asm volatile("s_wait_asynccnt 0x2" ::: "memory");
}
__device__ __forceinline__ void wait_async4() {
  asm volatile("s_wait_asynccnt 0x4" ::: "memory");
}

// ---------------------------------------------------------------------------
// LayerNorm (fp32, ddof=0, eps=1e-5): one block per row of 1024
// ---------------------------------------------------------------------------
__global__ __launch_bounds__(256) void ln_kernel(const float* __restrict__ x,
                                                 const float* __restrict__ g,
                                                 const float* __restrict__ b,
                                                 float* __restrict__ y) {
  __shared__ float rs[256];
  __shared__ float rq[256];
  const int tid = threadIdx.x;
  const float* xr = x + (size_t)blockIdx.x * D_MODEL;
  float4 xv = ((const float4*)xr)[tid];
  float s = xv.x + xv.y + xv.z + xv.w;
  float q = xv.x * xv.x + xv.y * xv.y + xv.z * xv.z + xv.w * xv.w;
  rs[tid] = s;
  rq[tid] = q;
  __syncthreads();
  for (int o = 128; o > 0; o >>= 1) {
    if (tid < o) {
      rs[tid] += rs[tid + o];
      rq[tid] += rq[tid + o];
    }
    __syncthreads();
  }
  const float mean = rs[0] * (1.0f / D_MODEL);
  const float var = rq[0] * (1.0f / D_MODEL) - mean * mean;
  const float inv = rsqrtf(var + 1e-5f);
  float4 gv = ((const float4*)g)[tid];
  float4 bv = ((const float4*)b)[tid];
  float4 yv;
  yv.x = (xv.x - mean) * inv * gv.x + bv.x;
  yv.y = (xv.y - mean) * inv * gv.y + bv.y;
  yv.z = (xv.z - mean) * inv * gv.z + bv.z;
  yv.w = (xv.w - mean) * inv * gv.w + bv.w;
  ((float4*)(y + (size_t)blockIdx.x * D_MODEL))[tid] = yv;
}

// ---------------------------------------------------------------------------
// LoRA down-proj: t[m][0..3] = h[m,:] . a0[r,:]  (+ rows 4..7 from a1 if set)
// ---------------------------------------------------------------------------
__global__ __launch_bounds__(256) void lora_t_kernel(
    const float* __restrict__ h, const float* __restrict__ a0,
    const float* __restrict__ a1, float* __restrict__ t, int tstride) {
  __shared__ float red[256];
  const int tid = threadIdx.x;
  const float* hr = h + (size_t)blockIdx.x * D_MODEL;
  float part[8];
#pragma unroll
  for (int r = 0; r < 8; ++r) part[r] = 0.0f;
  const int nr = a1 ? 8 : 4;
  for (int i = tid; i < D_MODEL; i += 256) {
    const float hv = hr[i];
#pragma unroll
    for (int r = 0; r < 4; ++r) part[r] += hv * a0[r * D_MODEL + i];
    if (a1) {
#pragma unroll
      for (int r = 0; r < 4; ++r) part[4 + r] += hv * a1[r * D_MODEL + i];
    }
  }
  for (int r = 0; r < nr; ++r) {
    red[tid] = part[r];
    __syncthreads();
    for (int o = 128; o > 0; o >>= 1) {
      if (tid < o) red[tid] += red[tid + o];
      __syncthreads();
    }
    if (tid == 0) t[(size_t)blockIdx.x * tstride + r] = red[0];
    __syncthreads();
  }
}

// ---------------------------------------------------------------------------
// Generic fp32 WMMA GEMM: out[M,N] = A[M,K] * W[N,K]^T (+bias +LoRA +GELU +res)
// 256 threads = 8 waves, 64x64 tile, BK=16, double-buffered async LDS staging.
// ---------------------------------------------------------------------------
__global__ __launch_bounds__(256) void gemm_wmma_kernel(
    const float* __restrict__ A, const float* __restrict__ W,
    const float* __restrict__ bias, const float* __restrict__ lb,
    const float* __restrict__ tmat, int tstride, int toff,
    const float* __restrict__ resid, float* __restrict__ out, int N, int K,
    int dogelu) {
  __shared__ float As[2][64][20];  // stride 20: conflict-free, 16B aligned
  __shared__ float Ws[2][64][20];
  const int tid = threadIdx.x;
  const int lane = tid & 31, wave = tid >> 5;
  const int wy = wave & 3, wx = wave >> 2;
  const int lane16 = lane & 15, half = lane >> 4;
  const int m0 = blockIdx.y * 64, n0 = blockIdx.x * 64;
  const int lrow = tid >> 2, lseg = (tid & 3) * 4;
  const float* Aload = A + (size_t)(m0 + lrow) * K + lseg;
  const float* Wload = W + (size_t)(n0 + lrow) * K + lseg;
  const unsigned aoff[2] = {lds_addr(&As[0][lrow][lseg]),
                            lds_addr(&As[1][lrow][lseg])};
  const unsigned woff[2] = {lds_addr(&Ws[0][lrow][lseg]),
                            lds_addr(&Ws[1][lrow][lseg])};

  v8f acc0 = {0, 0, 0, 0, 0, 0, 0, 0};
  v8f acc1 = {0, 0, 0, 0, 0, 0, 0, 0};

  // prologue: stage tile 0 into buffer 0
  async_ld16(aoff[0], Aload);
  async_ld16(woff[0], Wload);

  const int niter = K >> 4;
  for (int it = 0; it < niter; ++it) {
    const int buf = it & 1;
    if (it + 1 < niter) {  // stage next tile, keep it in flight
      const int nk = (it + 1) << 4;
      async_ld16(aoff[buf ^ 1], Aload + nk);
      async_ld16(woff[buf ^ 1], Wload + nk);
      wait_async2();  // in-order completion: tile `it` has landed
    } else {
      wait_async0();
    }
    __syncthreads();
#pragma unroll
    for (int s = 0; s < 4; ++s) {
      const int col = 4 * s + 2 * half;
      v2f a = *(const v2f*)&As[buf][16 * wy + lane16][col];
      v2f b0 = *(const v2f*)&Ws[buf][32 * wx + lane16][col];
      v2f b1 = *(const v2f*)&Ws[buf][32 * wx + 16 + lane16][col];
      acc0 = wmma_f32(a, b0, acc0);
      acc1 = wmma_f32(a, b1, acc1);
    }
    __syncthreads();  // all waves done with `buf` before it is re-staged
  }

  // Epilogue: C layout — VGPR j holds row (j + 8*half), col = lane16.
#pragma unroll
  for (int j = 0; j < 8; ++j) {
    const int row = m0 + 16 * wy + j + 8 * half;
#pragma unroll
    for (int t = 0; t < 2; ++t) {
      const int col = n0 + 32 * wx + 16 * t + lane16;
      float v = (t == 0) ? acc0[j] : acc1[j];
      if (bias) v += bias[col];
      if (lb) {  // rank-4 LoRA up-proj, scale = alpha/r = 0.25
        const float* tm = tmat + (size_t)row * tstride + toff;
        const float* lp = lb + (size_t)col * 4;
        v += 0.25f * (lp[0] * tm[0] + lp[1] * tm[1] + lp[2] * tm[2] +
                      lp[3] * tm[3]);
      }
      if (dogelu) v = 0.5f * v * (1.0f + erff(v * 0.70710678118654752f));
      if (resid) v += resid[(size_t)row * N + col];
      out[(size_t)row * N + col] = v;
    }
  }
}

// ---------------------------------------------------------------------------
// Flash attention, fp32 WMMA. Block = 128 threads = 4 waves; each wave owns a
// 16-row Q tile of one (b,h); K/V 16x64 chunks double-buffered via async LDS.
// scores = (Q*hd^-.25)(K*hd^-.25)^T == QK^T * 0.125 for hd=64.
// ---------------------------------------------------------------------------
__global__ __launch_bounds__(128) void attn_kernel(const float* __restrict__ q,
                                                   const float* __restrict__ k,
                                                   const float* __restrict__ v,
                                                   float* __restrict__ wv) {
  __shared__ float Ks[2][16][68];
  __shared__ float Vs[2][16][68];
  __shared__ float Ps[4][16][20];
  const int tid = threadIdx.x;
  const int lane = tid & 31, wave = tid >> 5;
  const int lane16 = lane & 15, half = lane >> 4;
  const int gid = blockIdx.x;
  const int qblk = gid & 15;  // SEQ/64 = 16 q-blocks
  const int bh = gid >> 4;
  const int hh = bh & 15, bb = bh >> 4;
  const int q0 = qblk * 64 + wave * 16;

  const float* Qbase = q + (size_t)(bb * SEQ) * D_MODEL + hh * HDIM;
  const float* Kbase = k + (size_t)(bb * SEQ) * D_MODEL + hh * HDIM;
  const float* Vbase = v + (size_t)(bb * SEQ) * D_MODEL + hh * HDIM;

  // Q fragments: 16 K-steps of 4 over hd=64
  v2f qa[16];
  {
    const float* qr = Qbase + (size_t)(q0 + lane16) * D_MODEL + 2 * half;
#pragma unroll
    for (int s = 0; s < 16; ++s) qa[s] = *(const v2f*)(qr + 4 * s);
  }

  v8f o0 = {0, 0, 0, 0, 0, 0, 0, 0}, o1 = {0, 0, 0, 0, 0, 0, 0, 0};
  v8f o2 = {0, 0, 0, 0, 0, 0, 0, 0}, o3 = {0, 0, 0, 0, 0, 0, 0, 0};
  float mrow[8], lrow[8];
#pragma unroll
  for (int j = 0; j < 8; ++j) {
    mrow[j] = -INFINITY;
    lrow[j] = 0.0f;
  }

  const int lr = tid >> 3;       // 0..15 stage row
  const int lc = (tid & 7) * 8;  // 0..56 stage col (two b128 per thread)
  const float* Kst = Kbase + (size_t)lr * D_MODEL + lc;
  const float* Vst = Vbase + (size_t)lr * D_MODEL + lc;
  const unsigned koff[2] = {lds_addr(&Ks[0][lr][lc]), lds_addr(&Ks[1][lr][lc])};
  const unsigned voff[2] = {lds_addr(&Vs[0][lr][lc]), lds_addr(&Vs[1][lr][lc])};

  // prologue: stage chunk 0 into buffer 0 (4 async ops)
  async_ld16(koff[0], Kst);
  async_ld16_o16(koff[0], Kst);
  async_ld16(voff[0], Vst);
  async_ld16_o16(voff[0], Vst);

  for (int it = 0; it < SEQ / 16; ++it) {
    const int buf = it & 1;
    if (it + 1 < SEQ / 16) {
      const float* kg = Kst + (size_t)(it + 1) * 16 * D_MODEL;
      const float* vg = Vst + (size_t)(it + 1) * 16 * D_MODEL;
      async_ld16(koff[buf ^ 1], kg);
      async_ld16_o16(koff[buf ^ 1], kg);
      async_ld16(voff[buf ^ 1], vg);
      async_ld16_o16(voff[buf ^ 1], vg);
      wait_async4();  // chunk `it` landed; chunk `it+1` still in flight
    } else {
      wait_async0();
    }
    __syncthreads();

    // scores: C[16 q, 16 keys]
    v8f c = {0, 0, 0, 0, 0, 0, 0, 0};
#pragma unroll
    for (int s = 0; s < 16; ++s) {
      v2f bf = *(const v2f*)&Ks[buf][lane16][4 * s + 2 * half];
      c = wmma_f32(qa[s], bf, c);
    }

    // online softmax; rows (j + 8*half) live in 16-lane halves
#pragma unroll
    for (int j = 0; j < 8; ++j) {
      float sc = c[j] * 0.125f;
      float rmax = sc;
      rmax = fmaxf(rmax, __shfl_xor(rmax, 1, 32));
      rmax = fmaxf(rmax, __shfl_xor(rmax, 2, 32));
      rmax = fmaxf(rmax, __shfl_xor(rmax, 4, 32));
      rmax = fmaxf(rmax, __shfl_xor(rmax, 8, 32));
      const float mn = fmaxf(mrow[j], rmax);
      const float f = expf(mrow[j] - mn);
      const float p = expf(sc - mn);
      float rsum = p;
      rsum += __shfl_xor(rsum, 1, 32);
      rsum += __shfl_xor(rsum, 2, 32);
      rsum += __shfl_xor(rsum, 4, 32);
      rsum += __shfl_xor(rsum, 8, 32);
      lrow[j] = lrow[j] * f + rsum;
      mrow[j] = mn;
      o0[j] *= f;
      o1[j] *= f;
      o2[j] *= f;
      o3[j] *= f;
      Ps[wave][j + 8 * half][lane16] = p;  // transpose C-layout -> A-layout
    }

    // PV: A = P(16x16), B = V chunk (16x64), 4 K-steps x 4 N-subtiles
#pragma unroll
    for (int s = 0; s < 4; ++s) {
      const int kk = 4 * s + 2 * half;
      v2f a = *(const v2f*)&Ps[wave][lane16][kk];
      v2f b0, b1, b2, b3;
      b0.x = Vs[buf][kk][lane16];
      b0.y = Vs[buf][kk + 1][lane16];
      b1.x = Vs[buf][kk][16 + lane16];
      b1.y = Vs[buf][kk + 1][16 + lane16];
      b2.x = Vs[buf][kk][32 + lane16];
      b2.y = Vs[buf][kk + 1][32 + lane16];
      b3.x = Vs[buf][kk][48 + lane16];
      b3.y = Vs[buf][kk + 1][48 + lane16];
      o0 = wmma_f32(a, b0, o0);
      o1 = wmma_f32(a, b1, o1);
      o2 = wmma_f32(a, b2, o2);
      o3 = wmma_f32(a, b3, o3);
    }
    __syncthreads();  // all waves done with `buf` before it is re-staged
  }

  float* wbase = wv + (size_t)(bb * SEQ) * D_MODEL + hh * HDIM;
#pragma unroll
  for (int j = 0; j < 8; ++j) {
    const int row = q0 + j + 8 * half;
    const float invl = 1.0f / lrow[j];
    float* wr = wbase + (size_t)row * D_MODEL;
    wr[lane16] = o0[j] * invl;
    wr[16 + lane16] = o1[j] * invl;
    wr[32 + lane16] = o2[j] * invl;
    wr[48 + lane16] = o3[j] * invl;
  }
}

// ---------------------------------------------------------------------------
extern "C" void kernel_launch(void* const* d_in, const int* in_sizes, int n_in,
                              void* d_out, int out_size, void* d_ws,
                              size_t ws_size, hipStream_t stream) {
  (void)in_sizes;
  (void)n_in;
  (void)out_size;
  (void)ws_size;
  const float* x = (const float*)d_in[0];
  const float* Wq = (const float*)d_in[1];
  const float* bq = (const float*)d_in[2];
  const float* Wk = (const float*)d_in[3];
  const float* Wv = (const float*)d_in[4];
  const float* bv = (const float*)d_in[5];
  const float* Wo = (const float*)d_in[6];
  const float* bo = (const float*)d_in[7];
  const float* la_k = (const float*)d_in[8];
  const float* lb_k = (const float*)d_in[9];
  const float* la_v = (const float*)d_in[10];
  const float* lb_v = (const float*)d_in[11];
  const float* la_o = (const float*)d_in[12];
  const float* lb_o = (const float*)d_in[13];
  const float* g_attn = (const float*)d_in[14];
  const float* b_attn = (const float*)d_in[15];
  const float* W1 = (const float*)d_in[16];
  const float* b1 = (const float*)d_in[17];
  const float* W2 = (const float*)d_in[18];
  const float* b2 = (const float*)d_in[19];
  const float* g_mlp = (const float*)d_in[20];
  const float* b_mlp = (const float*)d_in[21];
  float* out = (float*)d_out;

  char* ws = (char*)d_ws;
  const size_t SZ = (size_t)MTOT * D_MODEL * sizeof(float);  // 32 MB
  float* h = (float*)(ws);             // LN output (reused for LN2)
  float* qb = (float*)(ws + SZ);       // q
  float* kb = (float*)(ws + 2 * SZ);   // k
  float* vb = (float*)(ws + 3 * SZ);   // v
  float* wvb = (float*)(ws + 4 * SZ);  // attention output
  float* x1 = (float*)(ws + 5 * SZ);   // x + attn
  float* tkv = (float*)(ws + 6 * SZ);  // [M,8] lora k|v
  float* to = (float*)(ws + 6 * SZ + (size_t)MTOT * 8 * sizeof(float));
  float* ffn = qb;  // reuse contiguous q..wv region (128 MB) for fc1 output

  dim3 b256(256), b128(128);
  dim3 gD(D_MODEL / 64, MTOT / 64);
  dim3 gF(D_FF / 64, MTOT / 64);

  ln_kernel<<<MTOT, b256, 0, stream>>>(x, g_attn, b_attn, h);
  lora_t_kernel<<<MTOT, b256, 0, stream>>>(h, la_k, la_v, tkv, 8);
  gemm_wmma_kernel<<<gD, b256, 0, stream>>>(h, Wq, bq, nullptr, nullptr, 0, 0,
                                            nullptr, qb, D_MODEL, D_MODEL, 0);
  gemm_wmma_kernel<<<gD, b256, 0, stream>>>(h, Wk, nullptr, lb_k, tkv, 8, 0,
                                            nullptr, kb, D_MODEL, D_MODEL, 0);
  gemm_wmma_kernel<<<gD, b256, 0, stream>>>(h, Wv, bv, lb_v, tkv, 8, 4, nullptr,
                                            vb, D_MODEL, D_MODEL, 0);
  attn_kernel<<<BATCH * NHEAD * (SEQ / 64), b128, 0, stream>>>(qb, kb, vb, wvb);
  lora_t_kernel<<<MTOT, b256, 0, stream>>>(wvb, la_o, nullptr, to, 4);
  gemm_wmma_kernel<<<gD, b256, 0, stream>>>(wvb, Wo, bo, lb_o, to, 4, 0, x, x1,
                                            D_MODEL, D_MODEL, 0);
  ln_kernel<<<MTOT, b256, 0, stream>>>(x1, g_mlp, b_mlp, h);
  gemm_wmma_kernel<<<gF, b256, 0, stream>>>(h, W1, b1, nullptr, nullptr, 0, 0,
                                            nullptr, ffn, D_FF, D_MODEL, 1);
  gemm_wmma_kernel<<<gD, b256, 0, stream>>>(ffn, W2, b2, nullptr, nullptr, 0, 0,
                                            x1, out, D_MODEL, D_FF, 0);
}